// Q_ReLU_22823456211627
// MI455X (gfx1250) — compile-verified
//
#include <hip/hip_runtime.h>
#include <stdint.h>

#define NBIN 256

typedef float v4f __attribute__((ext_vector_type(4)));

// Deterministic counter-based uniform in [0,1) (PCG-style hash). Stands in
// for jax.random threefry streams; fully deterministic per element index.
__device__ __forceinline__ float u01_hash(uint32_t v) {
  v = v * 747796405u + 2891336453u;
  uint32_t w = ((v >> ((v >> 28u) + 4u)) ^ v) * 277803737u;
  w = (w >> 22u) ^ w;
  return (float)(w >> 8) * (1.0f / 16777216.0f);
}

// ---------------- pass 0: scalar params + zero histogram ----------------
__global__ void qrelu_params_kernel(const float* __restrict__ bit,
                                    const float* __restrict__ alpha,
                                    float* __restrict__ params,
                                    unsigned* __restrict__ hist) {
  hist[threadIdx.x] = 0u;  // blockDim.x == NBIN
  if (threadIdx.x == 0) {
    float bv  = bit[0];
    float sig = 1.0f / (1.0f + __expf(-bv));
    float b0  = 2.0f + sig * 12.0f;
    float u   = u01_hash(0x2545F491u);          // stochastic bit jitter
    float b   = rintf(b0 + u - 0.5f);           // STE forward value (RNE)
    float av  = alpha[0];
    float a   = (av > 20.0f) ? av : log1pf(__expf(av));  // softplus
    float n_lv  = exp2f(b) - 1.0f;
    float delta = a / n_lv;
    params[0] = a;
    params[1] = n_lv;
    params[2] = delta;
    params[3] = 1.0f / delta;
  }
}

// ---------------- pass 1: histogram of quantization error ----------------
__device__ __forceinline__ void hist_accum(float xv, float a, float n_lv,
                                           float inv_a, float delta,
                                           float inv_delta,
                                           unsigned* __restrict__ lh) {
  float dq   = fminf(fmaxf(xv * inv_a, 0.0f), 1.0f) * n_lv;
  float lsq  = rintf(dq) * delta;               // round half-to-even == jnp.round
  float diff = (lsq - xv) * inv_delta;
  bool mask  = (xv < a) && (xv > 0.0f);         // !(x>=a || x<=0)
  bool inr   = (diff >= -0.5f) && (diff <= 0.5f);
  if (mask && inr) {
    int idx = (int)floorf((diff + 0.5f) * (float)NBIN);
    idx = idx < 0 ? 0 : (idx > NBIN - 1 ? NBIN - 1 : idx);
    atomicAdd(&lh[idx], 1u);                    // ds_add_u32 (LDS atomic)
  }
}

__global__ void qrelu_hist_kernel(const float* __restrict__ x, long long n,
                                  const float* __restrict__ params,
                                  unsigned* __restrict__ hist) {
  __shared__ unsigned lh[NBIN];
  lh[threadIdx.x] = 0u;  // blockDim.x == NBIN
  __syncthreads();

  float a = params[0], n_lv = params[1], delta = params[2], inv_d = params[3];
  float inv_a = 1.0f / a;

  long long n4 = n >> 2;
  const v4f* __restrict__ x4 = (const v4f*)x;
  long long stride = (long long)gridDim.x * blockDim.x;
  for (long long i = (long long)blockIdx.x * blockDim.x + threadIdx.x; i < n4;
       i += stride) {
    __builtin_prefetch(&x4[i + stride], 0, 0);  // global_prefetch_b8
    v4f v = __builtin_nontemporal_load(&x4[i]); // b128 load, TH_NT (streamed)
    hist_accum(v.x, a, n_lv, inv_a, delta, inv_d, lh);
    hist_accum(v.y, a, n_lv, inv_a, delta, inv_d, lh);
    hist_accum(v.z, a, n_lv, inv_a, delta, inv_d, lh);
    hist_accum(v.w, a, n_lv, inv_a, delta, inv_d, lh);
  }
  for (long long i = (n4 << 2) + (long long)blockIdx.x * blockDim.x + threadIdx.x;
       i < n; i += stride) {
    hist_accum(x[i], a, n_lv, inv_a, delta, inv_d, lh);
  }

  __syncthreads();
  unsigned c = lh[threadIdx.x];
  if (c) atomicAdd(&hist[threadIdx.x], c);      // global_atomic_add_u32
}

// ---------------- pass 2: normalize + inclusive scan -> CDF ----------------
__global__ void qrelu_cdf_kernel(const unsigned* __restrict__ hist,
                                 float* __restrict__ cdf) {
  __shared__ float s[NBIN];
  int t = threadIdx.x;  // blockDim.x == NBIN
  s[t] = (float)hist[t];
  __syncthreads();
#pragma unroll
  for (int off = 1; off < NBIN; off <<= 1) {
    float v = (t >= off) ? s[t - off] : 0.0f;
    __syncthreads();
    s[t] += v;
    __syncthreads();
  }
  float total = s[NBIN - 1];
  cdf[t] = s[t] / fmaxf(total, 1.0f);
}

// ---------------- pass 3: sample noise + fused Q_ReLU output ----------------
__device__ __forceinline__ float qrelu_elem(float xv, uint32_t gi, float a,
                                            float delta,
                                            const float* __restrict__ scdf) {
  float us = u01_hash(gi * 2u + 0x9E3779B9u);   // selection uniform
  float uf = u01_hash(gi * 2u + 0x85EBCA77u);   // fractional uniform
  // searchsorted(cdf, us, side='right'): count of entries <= us, branchless.
  int pos = 0;
#pragma unroll
  for (int s = NBIN / 2; s > 0; s >>= 1)
    if (scdf[pos + s - 1] <= us) pos += s;      // ds_load_b32 gathers
  pos = pos > NBIN - 1 ? NBIN - 1 : pos;
  float noise = ((float)pos + uf) * (1.0f / (float)NBIN) - 0.5f;
  return (xv >= a) ? a : ((xv <= 0.0f) ? 0.0f : fmaf(noise, delta, xv));
}

__global__ void qrelu_out_kernel(const float* __restrict__ x,
                                 float* __restrict__ out, long long n,
                                 const float* __restrict__ params,
                                 const float* __restrict__ cdf) {
  __shared__ float scdf[NBIN];
  {
    // gfx1250 async-to-LDS: each of the 256 threads pulls 4B of the CDF
    // straight into LDS, bypassing VGPRs; low 32 bits of a flat shared
    // pointer are the LDS byte offset.
    uint32_t lds_addr = (uint32_t)(uintptr_t)(&scdf[threadIdx.x]);
    const float* gp = cdf + threadIdx.x;
    asm volatile("global_load_async_to_lds_b32 %0, %1, off"
                 :: "v"(lds_addr), "v"(gp) : "memory");
    asm volatile("s_wait_asynccnt 0" ::: "memory");
  }
  __syncthreads();

  float a = params[0];
  float delta = params[2];

  long long n4 = n >> 2;
  const v4f* __restrict__ x4 = (const v4f*)x;
  v4f* __restrict__ o4 = (v4f*)out;
  long long stride = (long long)gridDim.x * blockDim.x;
  for (long long i = (long long)blockIdx.x * blockDim.x + threadIdx.x; i < n4;
       i += stride) {
    v4f v = __builtin_nontemporal_load(&x4[i]); // b128 load, TH_NT
    uint32_t gi = (uint32_t)(i << 2);
    v4f r;
    r.x = qrelu_elem(v.x, gi + 0u, a, delta, scdf);
    r.y = qrelu_elem(v.y, gi + 1u, a, delta, scdf);
    r.z = qrelu_elem(v.z, gi + 2u, a, delta, scdf);
    r.w = qrelu_elem(v.w, gi + 3u, a, delta, scdf);
    __builtin_nontemporal_store(r, &o4[i]);     // b128 store, TH_NT (write-once)
  }
  for (long long i = (n4 << 2) + (long long)blockIdx.x * blockDim.x + threadIdx.x;
       i < n; i += stride) {
    out[i] = qrelu_elem(x[i], (uint32_t)i, a, delta, scdf);
  }
}

// ---------------- launcher ----------------
extern "C" void kernel_launch(void* const* d_in, const int* in_sizes, int n_in,
                              void* d_out, int out_size, void* d_ws,
                              size_t ws_size, hipStream_t stream) {
  const float* x     = (const float*)d_in[0];
  const float* bit   = (const float*)d_in[1];
  const float* alpha = (const float*)d_in[2];
  float* out = (float*)d_out;
  long long n = (long long)in_sizes[0];

  // Workspace layout: 64B params | 1KiB hist(u32) | 1KiB cdf(f32)
  float*    params = (float*)d_ws;
  unsigned* hist   = (unsigned*)((char*)d_ws + 64);
  float*    cdf    = (float*)((char*)d_ws + 64 + NBIN * sizeof(unsigned));

  const int BLOCK = NBIN;  // 256 threads = 8 wave32 waves
  long long n4   = n >> 2;
  long long want = (n4 + BLOCK - 1) / BLOCK;
  int grid = (int)(want < 1 ? 1 : (want > 4096 ? 4096 : want));

  qrelu_params_kernel<<<1, NBIN, 0, stream>>>(bit, alpha, params, hist);
  qrelu_hist_kernel<<<grid, BLOCK, 0, stream>>>(x, n, params, hist);
  qrelu_cdf_kernel<<<1, NBIN, 0, stream>>>(hist, cdf);
  qrelu_out_kernel<<<grid, BLOCK, 0, stream>>>(x, out, n, params, cdf);
}